// GraphSAGE_Node_44272522887304
// MI455X (gfx1250) — compile-verified
//
#include <hip/hip_runtime.h>
#include <hip/hip_bf16.h>

typedef float v2f __attribute__((ext_vector_type(2)));
typedef float v8f __attribute__((ext_vector_type(8)));

#define NNODES 100000
#define CIN 128

// ---------------------------------------------------------------- degree ----
__global__ void deg_count_kernel(const int* __restrict__ dst,
                                 float* __restrict__ deg, int E) {
    int e = blockIdx.x * blockDim.x + threadIdx.x;
    if (e < E) atomicAdd(&deg[dst[e]], 1.0f);
}

__global__ void deg_inv_kernel(float* __restrict__ deg, int N) {
    int n = blockIdx.x * blockDim.x + threadIdx.x;
    if (n < N) deg[n] = 1.0f / fmaxf(deg[n], 1.0f);
}

// --------------------------------------------------- edge gather/scatter ----
// idx = e*32 + g : thread g handles 4 channels [4g, 4g+4) of edge e.
// h rows (51.2MB) and agg rows (51.2MB) both live in the 192MB L2, so the
// float4 gather + 4x global_atomic_add_f32 run at L2 rates, not HBM.
__global__ void edge_scatter_kernel(const float* __restrict__ h,
                                    const int* __restrict__ src,
                                    const int* __restrict__ dst,
                                    float* __restrict__ agg, int E) {
    int idx = blockIdx.x * blockDim.x + threadIdx.x;
    if (idx >= E * 32) return;
    int e = idx >> 5;
    int g = idx & 31;
    int s = src[e];
    int d = dst[e];
    const float4 v = ((const float4*)(h + (size_t)s * CIN))[g];
    float* a = agg + (size_t)d * CIN + g * 4;
    atomicAdd(a + 0, v.x);
    atomicAdd(a + 1, v.y);
    atomicAdd(a + 2, v.z);
    atomicAdd(a + 3, v.w);
}

// ------------------------------------------------------------ fused GEMM ----
// out[n][o] = sum_k (agg[n][k]*dinv[n])*Wl[o][k] + bl[o] + sum_k h[n][k]*Wr[o][k]
// One block = one 16-node M-tile.  Wave w owns the 16x16 tile at n_base=16w.
// A' = [agg*dinv | h] (16x256) staged in LDS; K=256 swept in 64 chained
// v_wmma_f32_16x16x4_f32 ops (full f32 precision, matching the reference).
__global__ void sage_gemm_kernel(const float* __restrict__ agg,
                                 const float* __restrict__ hin,
                                 const float* __restrict__ dinv,
                                 const float* __restrict__ Wl,
                                 const float* __restrict__ bl,
                                 const float* __restrict__ Wr,
                                 float* __restrict__ out,
                                 int n_out, int do_relu) {
    __shared__ float lds[16][260];   // 260-float row stride: conflict-free, 16B aligned

    const int tid    = threadIdx.x;
    const int T      = blockDim.x;
    const int m_base = blockIdx.x * 16;

    // Stage A' tile: cols 0..127 = agg*dinv, cols 128..255 = h.  float4 granularity.
    for (int gidx = tid; gidx < 16 * 64; gidx += T) {
        int row  = gidx >> 6;
        int cg   = gidx & 63;
        int node = m_base + row;
        float4 v;
        if (cg < 32) {
            v = ((const float4*)(agg + (size_t)node * CIN))[cg];
            float s = dinv[node];
            v.x *= s; v.y *= s; v.z *= s; v.w *= s;
        } else {
            v = ((const float4*)(hin + (size_t)node * CIN))[cg - 32];
        }
        *(float4*)&lds[row][cg * 4] = v;
    }
    __syncthreads();

    const int wave   = tid >> 5;
    const int lane   = tid & 31;
    const int lrow   = lane & 15;   // A: M row | B/C/D: N column
    const int khalf  = lane >> 4;   // 0: K+{0,1}  1: K+{2,3}
    const int n_base = wave * 16;

    // C init = bias (depends on N only)
    const float bias = bl[n_base + lrow];
    v8f acc;
#pragma unroll
    for (int i = 0; i < 8; ++i) acc[i] = bias;

    // Part 1: (agg*dinv) x Wl^T   -- LDS cols [0,128)
    {
        const float* wb = Wl + (size_t)(n_base + lrow) * CIN;
#pragma unroll
        for (int k0 = 0; k0 < 128; k0 += 4) {
            int ka = k0 + khalf * 2;
            v2f a = *(const v2f*)&lds[lrow][ka];
            v2f b = *(const v2f*)(wb + ka);
            acc = __builtin_amdgcn_wmma_f32_16x16x4_f32(
                false, a, false, b, (short)0, acc, false, false);
        }
    }
    // Part 2: h x Wr^T   -- LDS cols [128,256)
    {
        const float* wb = Wr + (size_t)(n_base + lrow) * CIN;
#pragma unroll
        for (int k0 = 0; k0 < 128; k0 += 4) {
            int ka = k0 + khalf * 2;
            v2f a = *(const v2f*)&lds[lrow][128 + ka];
            v2f b = *(const v2f*)(wb + ka);
            acc = __builtin_amdgcn_wmma_f32_16x16x4_f32(
                false, a, false, b, (short)0, acc, false, false);
        }
    }

    if (do_relu) {
#pragma unroll
        for (int i = 0; i < 8; ++i) acc[i] = fmaxf(acc[i], 0.0f);
    }

    // D layout: VGPR v, lanes 0-15 -> (M=v, N=lrow); lanes 16-31 -> (M=v+8, N=lrow)
    const int mrow0 = khalf * 8;
#pragma unroll
    for (int v = 0; v < 8; ++v) {
        out[(size_t)(m_base + mrow0 + v) * n_out + n_base + lrow] = acc[v];
    }
}

// ------------------------------------------------------------------ host ----
extern "C" void kernel_launch(void* const* d_in, const int* in_sizes, int n_in,
                              void* d_out, int out_size, void* d_ws, size_t ws_size,
                              hipStream_t stream) {
    const float* x   = (const float*)d_in[0];
    // d_in[1] = edge_attr (unused by reference path)
    const int*   ei  = (const int*)d_in[2];
    const int    E   = in_sizes[2] / 2;
    const int*   src = ei;
    const int*   dst = ei + E;

    const float* Wl0 = (const float*)d_in[3];
    const float* bl0 = (const float*)d_in[4];
    const float* Wr0 = (const float*)d_in[5];
    const float* Wl1 = (const float*)d_in[6];
    const float* bl1 = (const float*)d_in[7];
    const float* Wr1 = (const float*)d_in[8];
    const float* Wl2 = (const float*)d_in[9];
    const float* bl2 = (const float*)d_in[10];
    const float* Wr2 = (const float*)d_in[11];

    float* out = (float*)d_out;

    // workspace layout
    float* ws  = (float*)d_ws;
    float* agg = ws;                                // NNODES*128
    float* h1  = agg + (size_t)NNODES * CIN;        // NNODES*128
    float* h2  = h1  + (size_t)NNODES * CIN;        // NNODES*128
    float* deg = h2  + (size_t)NNODES * CIN;        // NNODES  (becomes deg_inv)

    const int scatterBlocks = (E * 32 + 255) / 256;
    const int mTiles        = NNODES / 16;          // 100000/16 = 6250 exactly

    // degree (depends only on dst; compute once, invert in place)
    hipMemsetAsync(deg, 0, (size_t)NNODES * sizeof(float), stream);
    deg_count_kernel<<<(E + 255) / 256, 256, 0, stream>>>(dst, deg, E);
    deg_inv_kernel<<<(NNODES + 255) / 256, 256, 0, stream>>>(deg, NNODES);

    // layer 0: x -> h1 (ReLU)
    hipMemsetAsync(agg, 0, (size_t)NNODES * CIN * sizeof(float), stream);
    edge_scatter_kernel<<<scatterBlocks, 256, 0, stream>>>(x, src, dst, agg, E);
    sage_gemm_kernel<<<mTiles, 256, 0, stream>>>(agg, x, deg, Wl0, bl0, Wr0, h1, 128, 1);

    // layer 1: h1 -> h2 (ReLU)
    hipMemsetAsync(agg, 0, (size_t)NNODES * CIN * sizeof(float), stream);
    edge_scatter_kernel<<<scatterBlocks, 256, 0, stream>>>(h1, src, dst, agg, E);
    sage_gemm_kernel<<<mTiles, 256, 0, stream>>>(agg, h1, deg, Wl1, bl1, Wr1, h2, 128, 1);

    // layer 2: h2 -> out (no ReLU), n_out = 64 -> 4 waves / block
    hipMemsetAsync(agg, 0, (size_t)NNODES * CIN * sizeof(float), stream);
    edge_scatter_kernel<<<scatterBlocks, 256, 0, stream>>>(h2, src, dst, agg, E);
    sage_gemm_kernel<<<mTiles, 128, 0, stream>>>(agg, h2, deg, Wl2, bl2, Wr2, out, 64, 0);
}